// contrastBert_25881472926363
// MI455X (gfx1250) — compile-verified
//
#include <hip/hip_runtime.h>
#include <math.h>

typedef __attribute__((ext_vector_type(2))) float v2f;
typedef __attribute__((ext_vector_type(8))) float v8f;

#define WAVE 32

// ---------------------------------------------------------------------------
// Kernel 1: masked mean pooling over the sequence dimension (bandwidth-bound)
// pooled[n][h] = sum_s lhs[n][s][h]*mask[n][s] / sum_s mask[n][s]
// One block per hypothesis n, H/4 threads, float4 coalesced loads.
// ---------------------------------------------------------------------------
__global__ void pool_kernel(const float* __restrict__ lhs,
                            const float* __restrict__ mask,
                            float* __restrict__ pooled,
                            int S, int H) {
  const int n = blockIdx.x;
  const int tid = threadIdx.x;             // H/4 threads (H=768 -> 192)
  __shared__ float smask[1024];
  __shared__ float ssum;
  for (int i = tid; i < S; i += blockDim.x) smask[i] = mask[(size_t)n * S + i];
  __syncthreads();
  if (tid == 0) {
    float s = 0.f;
    for (int i = 0; i < S; ++i) s += smask[i];
    ssum = s;
  }
  __syncthreads();
  const float inv = 1.0f / ssum;
  const int stride4 = H >> 2;
  const float4* row = (const float4*)(lhs + (size_t)n * S * H) + tid;
  float4 acc = make_float4(0.f, 0.f, 0.f, 0.f);
  for (int s = 0; s < S; ++s) {
    float4 v = row[(size_t)s * stride4];
    if (s + 2 < S)
      __builtin_prefetch((const void*)(row + (size_t)(s + 2) * stride4), 0, 0);
    const float mk = smask[s];
    acc.x += v.x * mk; acc.y += v.y * mk; acc.z += v.z * mk; acc.w += v.w * mk;
  }
  float4* dst = (float4*)(pooled + (size_t)n * H) + tid;
  *dst = make_float4(acc.x * inv, acc.y * inv, acc.z * inv, acc.w * inv);
}

// ---------------------------------------------------------------------------
// Kernel 2: final_score[n] = pooler[n,:] . W[0:H] + am[n]*W[H] + ctc[n]*W[H+1] + b
// One wave32 per row; shuffle reduction.
// ---------------------------------------------------------------------------
__global__ void score_kernel(const float* __restrict__ P,
                             const float* __restrict__ am,
                             const float* __restrict__ ctc,
                             const float* __restrict__ W,
                             const float* __restrict__ bias,
                             float* __restrict__ score,
                             int N, int H) {
  const int wave = threadIdx.x / WAVE;
  const int lane = threadIdx.x % WAVE;
  const int n = blockIdx.x * (blockDim.x / WAVE) + wave;
  if (n >= N) return;
  const float* row = P + (size_t)n * H;
  float acc = 0.f;
  for (int j = lane; j < H; j += WAVE) acc += row[j] * W[j];
#pragma unroll
  for (int o = WAVE / 2; o > 0; o >>= 1) acc += __shfl_down(acc, o, WAVE);
  if (lane == 0)
    score[n] = acc + am[n] * W[H] + ctc[n] * W[H + 1] + bias[0];
}

// ---------------------------------------------------------------------------
// Kernel 3: segmented softmax + cross-entropy partial per n-best group.
// ce_partial[g] = sum_{i in g} labels[i] * (m + log(sum exp) - score[i])
// ---------------------------------------------------------------------------
__global__ void ce_kernel(const float* __restrict__ scores,
                          const float* __restrict__ labels,
                          const int* __restrict__ nbest,
                          float* __restrict__ ce_partial) {
  const int g = blockIdx.x;
  int off = 0;
  for (int i = 0; i < g; ++i) off += nbest[i];
  const int size = nbest[g];
  const int tid = threadIdx.x;
  __shared__ float red[256];

  float m = -INFINITY;
  for (int i = tid; i < size; i += blockDim.x) m = fmaxf(m, scores[off + i]);
  red[tid] = m; __syncthreads();
  for (int st = 128; st > 0; st >>= 1) {
    if (tid < st) red[tid] = fmaxf(red[tid], red[tid + st]);
    __syncthreads();
  }
  m = red[0]; __syncthreads();

  float s = 0.f;
  for (int i = tid; i < size; i += blockDim.x) s += __expf(scores[off + i] - m);
  red[tid] = s; __syncthreads();
  for (int st = 128; st > 0; st >>= 1) {
    if (tid < st) red[tid] += red[tid + st];
    __syncthreads();
  }
  const float logsum = __logf(red[0]); __syncthreads();

  float ce = 0.f;
  for (int i = tid; i < size; i += blockDim.x)
    ce += labels[off + i] * (m + logsum - scores[off + i]);
  red[tid] = ce; __syncthreads();
  for (int st = 128; st > 0; st >>= 1) {
    if (tid < st) red[tid] += red[tid + st];
    __syncthreads();
  }
  if (tid == 0) ce_partial[g] = red[0];
}

// ---------------------------------------------------------------------------
// Kernel 4: block-diagonal similarity via WMMA f32 16x16x4, then per-row
// softmax over the 32-column group, accumulate diagonal probabilities.
// One wave32 per group; 2x2 tiles of 16x16, K stepped by 4.
// A-lane layout (16x4 f32): lanes 0-15 hold K=k,k+1; lanes 16-31 hold K=k+2,k+3.
// B (4x16) layout is the mirror, so A and B loads share the same addressing.
// ---------------------------------------------------------------------------
__global__ void contrast_kernel(const float* __restrict__ P,
                                const float* __restrict__ pooled,
                                const int* __restrict__ nbest,
                                float* __restrict__ ct_partial,
                                int H) {
  const int g = blockIdx.x;
  int off = 0;
  for (int i = 0; i < g; ++i) off += nbest[i];
  const int size = nbest[g];
  const int lane = threadIdx.x;          // 0..31, single wave
  const int l15 = lane & 15;
  const int hi = lane >> 4;              // 0: K pair 0/1, 1: K pair 2/3
  __shared__ float Sblk[32 * 33];        // padded rows -> no LDS bank conflicts

  if (size == 32) {
    const float* Pa = P + (size_t)off * H;
    const float* Pb = pooled + (size_t)off * H;
    v8f c00 = {}, c01 = {}, c10 = {}, c11 = {};
    for (int k = 0; k < H; k += 4) {
      const int kk = k + hi * 2;
      v2f a0 = *(const v2f*)(Pa + (size_t)l15 * H + kk);
      v2f a1 = *(const v2f*)(Pa + (size_t)(l15 + 16) * H + kk);
      v2f b0 = *(const v2f*)(Pb + (size_t)l15 * H + kk);
      v2f b1 = *(const v2f*)(Pb + (size_t)(l15 + 16) * H + kk);
      c00 = __builtin_amdgcn_wmma_f32_16x16x4_f32(false, a0, false, b0,
                                                  (short)0, c00, false, false);
      c01 = __builtin_amdgcn_wmma_f32_16x16x4_f32(false, a0, false, b1,
                                                  (short)0, c01, false, false);
      c10 = __builtin_amdgcn_wmma_f32_16x16x4_f32(false, a1, false, b0,
                                                  (short)0, c10, false, false);
      c11 = __builtin_amdgcn_wmma_f32_16x16x4_f32(false, a1, false, b1,
                                                  (short)0, c11, false, false);
    }
    // C/D layout: VGPR r, lanes 0-15 -> row r, lanes 16-31 -> row r+8, col = lane&15
#pragma unroll
    for (int r = 0; r < 8; ++r) {
      const int row0 = hi * 8 + r;
      Sblk[row0 * 33 + l15]             = c00[r];
      Sblk[row0 * 33 + 16 + l15]        = c01[r];
      Sblk[(16 + row0) * 33 + l15]      = c10[r];
      Sblk[(16 + row0) * 33 + 16 + l15] = c11[r];
    }
    __syncthreads();
    // per-row softmax over 32 columns; keep the diagonal probability
    float m = -INFINITY;
#pragma unroll
    for (int c = 0; c < 32; ++c) m = fmaxf(m, Sblk[lane * 33 + c]);
    float s = 0.f;
#pragma unroll
    for (int c = 0; c < 32; ++c) s += __expf(Sblk[lane * 33 + c] - m);
    float diag = __expf(Sblk[lane * 33 + lane] - m) / s;
#pragma unroll
    for (int o = WAVE / 2; o > 0; o >>= 1) diag += __shfl_down(diag, o, WAVE);
    if (lane == 0) ct_partial[g] = diag;
  } else {
    // scalar fallback for irregular group sizes (size <= 32 supported)
    float diag = 0.f;
    if (lane < size && size <= 32) {
      for (int col = 0; col < size; ++col) {
        float dot = 0.f;
        for (int k = 0; k < H; ++k)
          dot += P[(size_t)(off + lane) * H + k] * pooled[(size_t)(off + col) * H + k];
        Sblk[lane * 33 + col] = dot;
      }
      float m = -INFINITY;
      for (int c = 0; c < size; ++c) m = fmaxf(m, Sblk[lane * 33 + c]);
      float s = 0.f;
      for (int c = 0; c < size; ++c) s += __expf(Sblk[lane * 33 + c] - m);
      diag = __expf(Sblk[lane * 33 + lane] - m) / s;
    }
#pragma unroll
    for (int o = WAVE / 2; o > 0; o >>= 1) diag += __shfl_down(diag, o, WAVE);
    if (lane == 0) ct_partial[g] = diag;
  }
}

// ---------------------------------------------------------------------------
// Kernel 5: loss = sum(ce_partial)/N - sum(ct_partial)/B
// ---------------------------------------------------------------------------
__global__ void finalize_kernel(const float* __restrict__ ce_partial,
                                const float* __restrict__ ct_partial,
                                float* __restrict__ loss,
                                int B, int N) {
  __shared__ float red[128];
  const int tid = threadIdx.x;
  float ce = 0.f, ct = 0.f;
  for (int i = tid; i < B; i += blockDim.x) {
    ce += ce_partial[i];
    ct += ct_partial[i];
  }
  red[tid] = ce; __syncthreads();
  for (int st = 64; st > 0; st >>= 1) {
    if (tid < st) red[tid] += red[tid + st];
    __syncthreads();
  }
  const float ceS = red[0]; __syncthreads();
  red[tid] = ct; __syncthreads();
  for (int st = 64; st > 0; st >>= 1) {
    if (tid < st) red[tid] += red[tid + st];
    __syncthreads();
  }
  if (tid == 0) loss[0] = ceS / (float)N - red[0] / (float)B;
}

// ---------------------------------------------------------------------------
extern "C" void kernel_launch(void* const* d_in, const int* in_sizes, int n_in,
                              void* d_out, int out_size, void* d_ws, size_t ws_size,
                              hipStream_t stream) {
  const float* pooler = (const float*)d_in[0];   // [N, H]
  const float* lhs    = (const float*)d_in[1];   // [N, S, H]
  const float* amask  = (const float*)d_in[2];   // [N, S]
  const float* am     = (const float*)d_in[3];   // [N, 1]
  const float* ctc    = (const float*)d_in[4];   // [N, 1]
  const float* labels = (const float*)d_in[5];   // [N]
  const float* W      = (const float*)d_in[6];   // [H+2, 1]
  const float* bias   = (const float*)d_in[7];   // [1]
  const int*   nbest  = (const int*)d_in[8];     // [B]

  const int H = in_sizes[6] - 2;                 // 768
  const int N = in_sizes[0] / H;                 // 4096
  const int S = in_sizes[2] / N;                 // 128
  const int B = in_sizes[8];                     // 128

  float* ws = (float*)d_ws;
  float* pooled      = ws;                       // N*H floats
  float* ce_partial  = ws + (size_t)N * H;       // B floats
  float* ct_partial  = ce_partial + B;           // B floats
  float* out = (float*)d_out;                    // [N] scores + [1] loss

  pool_kernel<<<N, H / 4, 0, stream>>>(lhs, amask, pooled, S, H);
  score_kernel<<<(N + 7) / 8, 256, 0, stream>>>(pooler, am, ctc, W, bias, out, N, H);
  ce_kernel<<<B, 256, 0, stream>>>(out, labels, nbest, ce_partial);
  contrast_kernel<<<B, WAVE, 0, stream>>>(pooler, pooled, nbest, ct_partial, H);
  finalize_kernel<<<1, 128, 0, stream>>>(ce_partial, ct_partial, out + N, B, N);
}